// UserRepeatCrossEntropyCriterion_65060164599829
// MI455X (gfx1250) — compile-verified
//
#include <hip/hip_runtime.h>
#include <hip/hip_bf16.h>
#include <math.h>

#define IGNORE_IDX (-100)
#define BS 4
#define SEQ 512
#define VOCAB 32000
#define CLAMP_MIN 1e-5f
#define THREADS 256
#define TILE 2048                               // floats per tile = 8 KB
#define NTILES ((VOCAB + TILE - 1) / TILE)      // 16 (last tile 1280 valid)
#define BM_WORDS ((VOCAB + 31) / 32)            // 1000 u32 per batch

#if defined(__HIP_DEVICE_COMPILE__) && \
    __has_builtin(__builtin_amdgcn_global_load_async_to_lds_b128) && \
    __has_builtin(__builtin_amdgcn_s_wait_asynccnt)
#define USE_ASYNC 1
#else
#define USE_ASYNC 0
#endif

struct Ws {
    double   acc;                    // combined mle + custom loss
    int      validCount;             // # of target_res != IGNORE
    int      pad_;
    int      ucount[BS];             // unique user tokens per batch
    unsigned bitmap[BS * BM_WORDS];  // vocab presence bitmap per batch
    int      ulist[BS * SEQ];        // compacted unique token ids
};

// ---------------------------------------------------------------------------
// Kernel A: zero accumulators, scatter user-token bitmap, compact to a list,
// count valid target_res. One block; phases separated by barriers.
// ---------------------------------------------------------------------------
__global__ __launch_bounds__(THREADS)
void urce_setup_kernel(const int* __restrict__ tu, const int* __restrict__ tr,
                       Ws* __restrict__ ws) {
    const int tid = threadIdx.x;
    if (tid == 0) { ws->acc = 0.0; ws->validCount = 0; }
    for (int i = tid; i < BS; i += THREADS) ws->ucount[i] = 0;
    for (int i = tid; i < BS * BM_WORDS; i += THREADS) ws->bitmap[i] = 0u;
    __syncthreads();

    int lv = 0;
    for (int i = tid; i < BS * SEQ; i += THREADS) {
        const int b = i / SEQ;
        const int t = tu[i];
        if (t != IGNORE_IDX)
            atomicOr(&ws->bitmap[b * BM_WORDS + (t >> 5)], 1u << (t & 31));
        lv += (tr[i] != IGNORE_IDX) ? 1 : 0;
    }
    atomicAdd(&ws->validCount, lv);
    __syncthreads();

    for (int w = tid; w < BS * BM_WORDS; w += THREADS) {
        const int b    = w / BM_WORDS;
        const int base = (w % BM_WORDS) * 32;
        unsigned bits  = ws->bitmap[w];
        while (bits) {
            const int bit = __ffs(bits) - 1;
            bits &= bits - 1u;
            const int pos = atomicAdd(&ws->ucount[b], 1);
            ws->ulist[b * SEQ + pos] = base + bit;   // <= 512 uniques per batch
        }
    }
}

// ---------------------------------------------------------------------------
// Kernel B: one block per (b,s) row. Async-stream the 32000-float row through
// a 2x8KB LDS double buffer, online softmax (m,d), then gather the unique
// user-token list for cl, and atomically add this row's loss contribution.
// ---------------------------------------------------------------------------
#if USE_ASYNC
typedef int v4i_t __attribute__((vector_size(16)));
typedef __attribute__((address_space(1))) void* as1_void_t;
typedef __attribute__((address_space(3))) void* as3_void_t;
typedef __attribute__((address_space(1))) v4i_t* as1_v4i_t;
typedef __attribute__((address_space(3))) v4i_t* as3_v4i_t;

__device__ __forceinline__ void async_cp_b128(const float* gsrc, float* ldst) {
    __builtin_amdgcn_global_load_async_to_lds_b128(
        (as1_v4i_t)(as1_void_t)(void*)gsrc,
        (as3_v4i_t)(as3_void_t)(void*)ldst,
        0, 0);
}
#endif

__global__ __launch_bounds__(THREADS)
void urce_row_kernel(const float* __restrict__ logits,
                     const int*  __restrict__ target_res,
                     const int*  __restrict__ belief_end,
                     Ws* __restrict__ ws) {
    const int row = blockIdx.x;          // 0 .. BS*SEQ-1
    const int b   = row / SEQ;
    const int s   = row % SEQ;
    const int tid = threadIdx.x;
    const float* __restrict__ g = logits + (size_t)row * VOCAB;

    __shared__ float red_m[THREADS];
    __shared__ float red_d[THREADS];
    __shared__ float s_logZ;

    float m = -INFINITY, d = 0.0f;

#if USE_ASYNC
    __shared__ float buf[2][TILE];
    // Each wave issues exactly 2 async b128 loads per tile (tail lanes clamp
    // to an in-row address; garbage region of the LDS buffer is never read).
    #define URCE_ISSUE(t_)                                                      \
        do {                                                                    \
            float* dst_ = buf[(t_) & 1];                                        \
            _Pragma("unroll")                                                   \
            for (int k_ = 0; k_ < 2; ++k_) {                                    \
                const int idx4_ = (k_ * THREADS + tid) * 4;                     \
                int gofs_ = (t_) * TILE + idx4_;                                \
                if (gofs_ > VOCAB - 4) gofs_ = VOCAB - 4;                       \
                async_cp_b128(g + gofs_, dst_ + idx4_);                         \
            }                                                                   \
        } while (0)

    URCE_ISSUE(0);
    URCE_ISSUE(1);
    for (int t = 0; t < NTILES; ++t) {
        if (t + 1 < NTILES) __builtin_amdgcn_s_wait_asynccnt(2);
        else                __builtin_amdgcn_s_wait_asynccnt(0);
        __syncthreads();                       // tile t visible from all waves
        const float* bp  = buf[t & 1];
        const int valid  = min(TILE, VOCAB - t * TILE);
        for (int j = tid; j < valid; j += THREADS) {
            const float x = bp[j];
            const float M = fmaxf(m, x);
            d = d * __expf(m - M) + __expf(x - M);
            m = M;
        }
        __syncthreads();                       // everyone done with buf[t&1]
        if (t + 2 < NTILES) URCE_ISSUE(t + 2); // refill the freed buffer
    }
    #undef URCE_ISSUE
#else
    for (int j = tid; j < VOCAB; j += THREADS) {
        const float x = g[j];
        const float M = fmaxf(m, x);
        d = d * __expf(m - M) + __expf(x - M);
        m = M;
    }
#endif

    // Block reduction of (m, d) pairs.
    red_m[tid] = m;
    red_d[tid] = d;
    __syncthreads();
    for (int off = THREADS / 2; off > 0; off >>= 1) {
        if (tid < off) {
            const float m1 = red_m[tid], d1 = red_d[tid];
            const float m2 = red_m[tid + off], d2 = red_d[tid + off];
            const float M  = fmaxf(m1, m2);
            red_d[tid] = d1 * __expf(m1 - M) + d2 * __expf(m2 - M);
            red_m[tid] = M;
        }
        __syncthreads();
    }
    if (tid == 0) s_logZ = red_m[0] + __logf(red_d[0]);
    __syncthreads();
    const float logZ = s_logZ;

    // cl = sum over unique user tokens of max(softmax_prob, 1e-5).
    const int c = ws->ucount[b];
    float part = 0.0f;
    for (int i = tid; i < c; i += THREADS) {
        const int tok = ws->ulist[b * SEQ + i];
        part += fmaxf(__expf(g[tok] - logZ), CLAMP_MIN);
    }
    red_d[tid] = part;
    __syncthreads();
    for (int off = THREADS / 2; off > 0; off >>= 1) {
        if (tid < off) red_d[tid] += red_d[tid + off];
        __syncthreads();
    }

    if (tid == 0) {
        double contrib = 0.0;
        const int trv = target_res[row];
        if (trv != IGNORE_IDX)
            contrib += (double)(logZ - g[trv]);            // -log p(target)
        if (s >= belief_end[b] && c > 0)                   // cl>0 iff c>0
            contrib += (double)(-__logf(red_d[0]));
        atomicAdd(&ws->acc, contrib);
    }
}

// ---------------------------------------------------------------------------
// Kernel C: scalar finalize.
// ---------------------------------------------------------------------------
__global__ void urce_finalize_kernel(const Ws* __restrict__ ws,
                                     float* __restrict__ out) {
    out[0] = (float)(ws->acc / (double)ws->validCount);
}

extern "C" void kernel_launch(void* const* d_in, const int* in_sizes, int n_in,
                              void* d_out, int out_size, void* d_ws, size_t ws_size,
                              hipStream_t stream) {
    const float* logits     = (const float*)d_in[0];
    const int*   target_usr = (const int*)d_in[1];
    const int*   target_res = (const int*)d_in[2];
    const int*   belief_end = (const int*)d_in[3];
    // d_in[4] (res_end) is unused by the reference.
    Ws*    ws  = (Ws*)d_ws;
    float* out = (float*)d_out;

    urce_setup_kernel<<<1, THREADS, 0, stream>>>(target_usr, target_res, ws);
    urce_row_kernel<<<BS * SEQ, THREADS, 0, stream>>>(logits, target_res,
                                                      belief_end, ws);
    urce_finalize_kernel<<<1, 1, 0, stream>>>(ws, out);
}